// OTKGE_wn_12481174962441
// MI455X (gfx1250) — compile-verified
//
#include <hip/hip_runtime.h>
#include <math.h>

// ---------------- problem constants ----------------
#define N_ENT   40943
#define RANK    256
#define TWO_RANK 512
#define IMG_DIM 1000
#define LING_DIM 768
#define BATCH   1000
#define NUM_OT  10
#define SK_ITERS 100
#define ALPHA   0.1f
#define GAMMA   0.8f
#define SCALE_C 100.0f
#define REG_C   0.05f

typedef __attribute__((ext_vector_type(2))) float v2f;
typedef __attribute__((ext_vector_type(4))) float v4f;
typedef __attribute__((ext_vector_type(8))) float v8f;

// ---------------- gfx1250 async LDS staging (guarded) ----------------
// Builtin prototype (from clang diagnostic): first arg is
// 'int __vector_size__(16) __device__ *'  ==  v4i addrspace(1)*;
// second is the LDS-side v4i addrspace(3)*; then imm offset, imm cpol.
typedef int async_v4i __attribute__((vector_size(16)));

#if defined(__AMDGCN__) && __has_builtin(__builtin_amdgcn_global_load_async_to_lds_b128)
#define HAVE_ASYNC_LDS 1
#define ASYNC_COPY16(gsrc, ldst)                                               \
    __builtin_amdgcn_global_load_async_to_lds_b128(                            \
        (__attribute__((address_space(1))) async_v4i*)(gsrc),                  \
        (__attribute__((address_space(3))) async_v4i*)(ldst), 0, 0)
#else
#define HAVE_ASYNC_LDS 0
#define ASYNC_COPY16(gsrc, ldst) (*(v4f*)(ldst) = *(const v4f*)(gsrc))
#endif

__device__ __forceinline__ void async_wait0() {
#if __has_builtin(__builtin_amdgcn_s_wait_asynccnt)
    __builtin_amdgcn_s_wait_asynccnt(0);
#else
    asm volatile("s_wait_asynccnt 0" ::: "memory");
#endif
}

// ---------------- workspace layout (float offsets) ----------------
// M1T/M2T stored TRANSPOSED: [TWO_RANK][K] (k-major per output column) so the
// per-lane WMMA B fragment (K=kk,kk+1 @ fixed column) is one contiguous v2f.
constexpr size_t WS_M1  = 0;                                    // [512][IMG_DIM]
constexpr size_t WS_M2  = WS_M1 + (size_t)TWO_RANK * IMG_DIM;   // [512][LING_DIM]
constexpr size_t WS_EMB = WS_M2 + (size_t)TWO_RANK * LING_DIM;  // [N_ENT][512]
constexpr size_t WS_P   = WS_EMB + (size_t)N_ENT * TWO_RANK;    // [BATCH][RANK]
constexpr size_t WS_Q   = WS_P + (size_t)BATCH * RANK;          // [BATCH][RANK]

// ============ kernel 1: M1T/M2T <- delta_ot (transposed copy) ============
__global__ void otkge_init_mats(const float* __restrict__ mats_img,
                                const float* __restrict__ mats_ling,
                                float* __restrict__ ws) {
    int i = blockIdx.x * blockDim.x + threadIdx.x;
    const int n1 = IMG_DIM * TWO_RANK;
    const int n2 = LING_DIM * TWO_RANK;
    if (i < n1) {
        const int k = i / TWO_RANK, n = i % TWO_RANK;       // coalesced read
        ws[WS_M1 + (size_t)n * IMG_DIM + k] = mats_img[i];
    } else if (i < n1 + n2) {
        const int j = i - n1;
        const int k = j / TWO_RANK, n = j % TWO_RANK;
        ws[WS_M2 + (size_t)n * LING_DIM + k] = mats_ling[j];
    }
}

// ============ kernel 2: log-domain Sinkhorn, one block per plan ============
__global__ void __launch_bounds__(512)
otkge_sinkhorn(const float* __restrict__ img_vec,
               const float* __restrict__ ling_vec,
               const float* __restrict__ emb_ent,
               float* __restrict__ ws) {
    const int blk = blockIdx.x;        // 0..19
    const int mdl = blk / NUM_OT;      // 0 = img, 1 = ling
    const int p   = blk % NUM_OT;
    const int mm_dim = (mdl == 0) ? IMG_DIM : LING_DIM;
    const float* mm_row = (mdl == 0) ? img_vec  + (size_t)p * IMG_DIM
                                     : ling_vec + (size_t)p * LING_DIM;
    const float* st_row = emb_ent + (size_t)p * TWO_RANK;
    float* outm = (mdl == 0) ? ws + WS_M1 : ws + WS_M2;   // transposed [512][mm_dim]

    __shared__ float s_mm[IMG_DIM];
    __shared__ float s_st[TWO_RANK];
    __shared__ float s_f[IMG_DIM];
    __shared__ float s_g[TWO_RANK];

    const int tid = threadIdx.x;
    for (int i = tid; i < mm_dim; i += 512) { s_mm[i] = mm_row[i]; s_f[i] = 0.f; }
    if (tid < TWO_RANK) { s_st[tid] = st_row[tid]; s_g[tid] = 0.f; }
    __syncthreads();

    const float loga = -logf((float)mm_dim);
    const float logb = -logf((float)TWO_RANK);
    const float inv_reg = 1.0f / REG_C;

    for (int it = 0; it < SK_ITERS; ++it) {
        // f[i] = REG*(loga - lse_j((g[j] - M_ij)/REG))
        for (int i = tid; i < mm_dim; i += 512) {
            const float mi = s_mm[i];
            float mx = -3.4e38f;
            for (int j = 0; j < TWO_RANK; ++j) {
                float d = s_st[j] - mi;
                float t = (s_g[j] - SCALE_C * d * d) * inv_reg;
                mx = fmaxf(mx, t);
            }
            float s = 0.f;
            for (int j = 0; j < TWO_RANK; ++j) {
                float d = s_st[j] - mi;
                float t = (s_g[j] - SCALE_C * d * d) * inv_reg;
                s += expf(t - mx);
            }
            s_f[i] = REG_C * (loga - (mx + logf(s)));
        }
        __syncthreads();
        // g[j] = REG*(logb - lse_i((f[i] - M_ij)/REG))
        if (tid < TWO_RANK) {
            const float sj = s_st[tid];
            float mx = -3.4e38f;
            for (int i = 0; i < mm_dim; ++i) {
                float d = sj - s_mm[i];
                float t = (s_f[i] - SCALE_C * d * d) * inv_reg;
                mx = fmaxf(mx, t);
            }
            float s = 0.f;
            for (int i = 0; i < mm_dim; ++i) {
                float d = sj - s_mm[i];
                float t = (s_f[i] - SCALE_C * d * d) * inv_reg;
                s += expf(t - mx);
            }
            s_g[tid] = REG_C * (logb - (mx + logf(s)));
        }
        __syncthreads();
    }

    // M_T[j][i] += mean(plan)*st_dim*SCALE  (w = 512*100/10 = 5120 per plan)
    const float w = (float)TWO_RANK * SCALE_C / (float)NUM_OT;
    const int total = mm_dim * TWO_RANK;
    for (int idx = tid; idx < total; idx += 512) {
        const int i = idx >> 9;      // row in cost matrix (mm index)
        const int j = idx & 511;     // st index (output column)
        float d = s_st[j] - s_mm[i];
        float t = (s_f[i] + s_g[j] - SCALE_C * d * d) * inv_reg;
        atomicAdd(&outm[(size_t)j * mm_dim + i], w * expf(t));
    }
}

// ============ kernel 3: embedding = c0*emb_ent + a*(img@M1) + g*(ling@M2) ====
// Block: 512 threads (16 waves), 16 entity rows/block. A-tile async-staged in
// LDS (img phase then ling phase reuse one <64KB buffer). Each wave owns two
// 16-wide N-tiles; one A fragment per k-step feeds both (named accumulators,
// no dynamic VGPR indexing). B fragments are contiguous v2f from M1T/M2T.
__global__ void __launch_bounds__(512)
otkge_embed_gemm(const float* __restrict__ emb_ent,
                 const float* __restrict__ img_vec,
                 const float* __restrict__ ling_vec,
                 float* __restrict__ ws) {
    constexpr int LDA = IMG_DIM + 8;       // 1008 floats; 16*1008*4 = 64512 B
    __shared__ float sA[16 * LDA];

    const int row0 = blockIdx.x * 16;
    const int tid  = threadIdx.x;
    const int wave = tid >> 5;
    const int lane = tid & 31;
    const int m    = lane & 15;   // A row / B column lane within tile
    const int kg   = lane >> 4;   // k-subgroup 0/1 (lane holds K = k0+2*kg+{0,1})

    const float* M1T = ws + WS_M1;
    const float* M2T = ws + WS_M2;
    float*       EMB = ws + WS_EMB;

    // B column indices for this lane's two N-tiles
    const int nA = (wave * 2 + 0) * 16 + m;
    const int nB = (wave * 2 + 1) * 16 + m;
    const float* bA1 = M1T + (size_t)nA * IMG_DIM;
    const float* bB1 = M1T + (size_t)nB * IMG_DIM;
    const float* bA2 = M2T + (size_t)nA * LING_DIM;
    const float* bB2 = M2T + (size_t)nB * LING_DIM;

    // Warm L2 for this lane's (L2-resident) B columns.
    __builtin_prefetch(bA1, 0, 1);
    __builtin_prefetch(bA2, 0, 1);

    v8f acc_i0 = (v8f){}, acc_i1 = (v8f){};
    v8f acc_l0 = (v8f){}, acc_l1 = (v8f){};

    // Clamped source row: rows >= N_ENT produce garbage that is never stored.
    const int e  = row0 + wave;
    const int ec = (e < N_ENT) ? e : (N_ENT - 1);

    // ---- phase 1: img_vec tile -> LDS (row = wave, b128 lane-strided) ----
    {
        const float* src = img_vec + (size_t)ec * IMG_DIM;
        for (int c = lane; c < IMG_DIM / 4; c += 32)            // 250 x 16B
            ASYNC_COPY16(src + c * 4, sA + wave * LDA + c * 4);
    }
#if HAVE_ASYNC_LDS
    async_wait0();
#endif
    __syncthreads();

    #pragma unroll 2
    for (int k0 = 0; k0 < IMG_DIM; k0 += 4) {
        const int kk = k0 + 2 * kg;
        v2f a  = *(const v2f*)(sA + m * LDA + kk);
        v2f b0 = *(const v2f*)(bA1 + kk);
        v2f b1 = *(const v2f*)(bB1 + kk);
        acc_i0 = __builtin_amdgcn_wmma_f32_16x16x4_f32(
            false, a, false, b0, (short)0, acc_i0, false, false);
        acc_i1 = __builtin_amdgcn_wmma_f32_16x16x4_f32(
            false, a, false, b1, (short)0, acc_i1, false, false);
    }
    __syncthreads();

    // ---- phase 2: ling_vec tile -> same LDS buffer ----
    {
        const float* src = ling_vec + (size_t)ec * LING_DIM;
        for (int c = lane; c < LING_DIM / 4; c += 32)           // 192 x 16B
            ASYNC_COPY16(src + c * 4, sA + wave * LDA + c * 4);
    }
#if HAVE_ASYNC_LDS
    async_wait0();
#endif
    __syncthreads();

    #pragma unroll 2
    for (int k0 = 0; k0 < LING_DIM; k0 += 4) {
        const int kk = k0 + 2 * kg;
        v2f a  = *(const v2f*)(sA + m * LDA + kk);
        v2f b0 = *(const v2f*)(bA2 + kk);
        v2f b1 = *(const v2f*)(bB2 + kk);
        acc_l0 = __builtin_amdgcn_wmma_f32_16x16x4_f32(
            false, a, false, b0, (short)0, acc_l0, false, false);
        acc_l1 = __builtin_amdgcn_wmma_f32_16x16x4_f32(
            false, a, false, b1, (short)0, acc_l1, false, false);
    }

    // ---- combine + store.  D layout: VGPR v -> M = kg*8 + v, N = tile + m ----
    const float c0 = 1.0f - ALPHA - GAMMA;
    #pragma unroll
    for (int v = 0; v < 8; ++v) {
        const int eo = row0 + kg * 8 + v;
        if (eo < N_ENT) {
            EMB[(size_t)eo * TWO_RANK + nA] =
                c0 * emb_ent[(size_t)eo * TWO_RANK + nA]
                + ALPHA * acc_i0[v] + GAMMA * acc_l0[v];
            EMB[(size_t)eo * TWO_RANK + nB] =
                c0 * emb_ent[(size_t)eo * TWO_RANK + nB]
                + ALPHA * acc_i1[v] + GAMMA * acc_l1[v];
        }
    }
}

// ============ kernel 4: gather batch, factors -> d_out, build P/Q ============
__global__ void __launch_bounds__(256)
otkge_prepare(const int* __restrict__ x,
              const float* __restrict__ emb_rel,
              float* __restrict__ ws,
              float* __restrict__ out) {
    const int b = blockIdx.x;
    const int k = threadIdx.x;
    const float* EMB = ws + WS_EMB;
    const int h = x[b * 3 + 0];
    const int r = x[b * 3 + 1];
    const int t = x[b * 3 + 2];
    const float l0 = EMB[(size_t)h * TWO_RANK + k];
    const float l1 = EMB[(size_t)h * TWO_RANK + RANK + k];
    const float r0 = emb_rel[(size_t)r * TWO_RANK + k];
    const float r1 = emb_rel[(size_t)r * TWO_RANK + RANK + k];
    const float h0 = EMB[(size_t)t * TWO_RANK + k];
    const float h1 = EMB[(size_t)t * TWO_RANK + RANK + k];
    ws[WS_P + (size_t)b * RANK + k] = l0 * r0 - l1 * r1;
    ws[WS_Q + (size_t)b * RANK + k] = l0 * r1 + l1 * r0;
    const size_t fb = (size_t)BATCH * N_ENT;
    out[fb + (size_t)b * RANK + k]                              = sqrtf(l0 * l0 + l1 * l1);
    out[fb + (size_t)BATCH * RANK + (size_t)b * RANK + k]       = sqrtf(r0 * r0 + r1 * r1);
    out[fb + 2 * (size_t)BATCH * RANK + (size_t)b * RANK + k]   = sqrtf(h0 * h0 + h1 * h1);
}

// ============ kernel 5: scores = P @ t0^T + Q @ t1^T  -> d_out[0:B*N] ========
// Block: 256 threads (8 waves). blockIdx.y -> 16 batch rows, blockIdx.x*8+wave
// -> entity 16-tile. P/Q rows async-staged in LDS (clamped, branchless);
// embedding columns from L2 as contiguous v2f.
__global__ void __launch_bounds__(256)
otkge_scores_gemm(const float* __restrict__ ws, float* __restrict__ out) {
    constexpr int LDP = RANK + 4;   // 260 floats; row stride 1040 B (16B-aligned)
    __shared__ float sP[16 * LDP];
    __shared__ float sQ[16 * LDP];

    const int b0  = blockIdx.y * 16;
    const int tid = threadIdx.x;
    const int wave = tid >> 5;
    const int lane = tid & 31;
    const int m    = lane & 15;
    const int kg   = lane >> 4;

    const float* P   = ws + WS_P;
    const float* Q   = ws + WS_Q;
    const float* EMB = ws + WS_EMB;

    // Fill: each wave stages rows {wave, wave+8}; rows >= BATCH clamped
    // (their outputs are never stored).
    for (int rr = wave; rr < 16; rr += 8) {
        const int b  = b0 + rr;
        const int bc = (b < BATCH) ? b : (BATCH - 1);
        const float* ps = P + (size_t)bc * RANK;
        const float* qs = Q + (size_t)bc * RANK;
        for (int c = lane; c < RANK / 4; c += 32) {             // 64 x 16B
            ASYNC_COPY16(ps + c * 4, sP + rr * LDP + c * 4);
            ASYNC_COPY16(qs + c * 4, sQ + rr * LDP + c * 4);
        }
    }
#if HAVE_ASYNC_LDS
    async_wait0();
#endif
    __syncthreads();

    const int NT = (N_ENT + 15) / 16;              // 2559 entity tiles
    const int et = blockIdx.x * 8 + wave;          // uniform per wave
    if (et >= NT) return;                          // whole wave exits: EXEC all-1s
    const int n0 = et * 16;
    const int n  = n0 + m;
    const int nc = (n < N_ENT) ? n : (N_ENT - 1);  // clamp; OOB col never stored
    const float* ecol = EMB + (size_t)nc * TWO_RANK;

    v8f acc = (v8f){};
    #pragma unroll 2
    for (int k0 = 0; k0 < RANK; k0 += 4) {
        const int kk = k0 + 2 * kg;
        v2f a, b;
        a = *(const v2f*)(sP + m * LDP + kk);
        b = *(const v2f*)(ecol + kk);
        acc = __builtin_amdgcn_wmma_f32_16x16x4_f32(
            false, a, false, b, (short)0, acc, false, false);
        a = *(const v2f*)(sQ + m * LDP + kk);
        b = *(const v2f*)(ecol + RANK + kk);
        acc = __builtin_amdgcn_wmma_f32_16x16x4_f32(
            false, a, false, b, (short)0, acc, false, false);
    }

    #pragma unroll
    for (int v = 0; v < 8; ++v) {
        const int mb = b0 + kg * 8 + v;
        if (mb < BATCH && n < N_ENT)
            out[(size_t)mb * N_ENT + n] = acc[v];
    }
}

// ============================ launcher ============================
extern "C" void kernel_launch(void* const* d_in, const int* in_sizes, int n_in,
                              void* d_out, int out_size, void* d_ws, size_t ws_size,
                              hipStream_t stream) {
    const int*   x         = (const int*)  d_in[0];
    const float* emb_ent   = (const float*)d_in[1];
    const float* emb_rel   = (const float*)d_in[2];
    const float* mats_img  = (const float*)d_in[3];
    const float* mats_ling = (const float*)d_in[4];
    const float* img_vec   = (const float*)d_in[5];
    const float* ling_vec  = (const float*)d_in[6];
    float* ws  = (float*)d_ws;
    float* out = (float*)d_out;

    // 1) M1T/M2T = delta_ot (transposed)
    const int n_init = IMG_DIM * TWO_RANK + LING_DIM * TWO_RANK;
    otkge_init_mats<<<(n_init + 255) / 256, 256, 0, stream>>>(mats_img, mats_ling, ws);

    // 2) Sinkhorn plans accumulate into M1T/M2T (20 plans)
    otkge_sinkhorn<<<2 * NUM_OT, 512, 0, stream>>>(img_vec, ling_vec, emb_ent, ws);

    // 3) fused multimodal embedding GEMM (WMMA f32)
    otkge_embed_gemm<<<(N_ENT + 15) / 16, 512, 0, stream>>>(emb_ent, img_vec, ling_vec, ws);

    // 4) batch gather, factors, P/Q
    otkge_prepare<<<BATCH, RANK, 0, stream>>>(x, emb_rel, ws, out);

    // 5) scores GEMM (WMMA f32)
    dim3 grid((((N_ENT + 15) / 16) + 7) / 8, (BATCH + 15) / 16);
    otkge_scores_gemm<<<grid, 256, 0, stream>>>(ws, out);
}